// GCN_27462020891318
// MI455X (gfx1250) — compile-verified
//
#include <hip/hip_runtime.h>
#include <math.h>

// Problem constants (from reference)
#define NU_N 100000
#define NI_N 50000
#define NN   150000
#define EE   2000000

typedef __attribute__((ext_vector_type(2))) float v2f;
typedef __attribute__((ext_vector_type(8))) float v8f;

// ---------------------------------------------------------------------------
// degree / dinv
// ---------------------------------------------------------------------------
__global__ void init_deg_kernel(float* deg, int n) {
  int i = blockIdx.x * blockDim.x + threadIdx.x;
  if (i < n) deg[i] = 1.0f;  // self-loop
}

__global__ void deg_count_kernel(const int* __restrict__ col, float* deg, int e) {
  int i = blockIdx.x * blockDim.x + threadIdx.x;
  if (i < e) atomicAdd(&deg[col[i]], 1.0f);
}

__global__ void deg_rsqrt_kernel(float* deg, int n) {
  int i = blockIdx.x * blockDim.x + threadIdx.x;
  if (i < n) deg[i] = rsqrtf(deg[i]);
}

__global__ void zero_kernel(float* p, int n) {
  int i = blockIdx.x * blockDim.x + threadIdx.x;
  if (i < n) p[i] = 0.0f;
}

// ---------------------------------------------------------------------------
// Dense GEMM  Y[N,64] = X[N,KIN] @ W[KIN,64]   via V_WMMA_F32_16X16X4_F32.
// One wave -> 16 rows x 64 cols (4 accumulators). wave32; EXEC all-ones.
// A layout (16x4 f32): lanes 0-15 row M=l16 K={k,k+1}; lanes 16-31 K={k+2,k+3}.
// B layout (4x16 f32): mirrored: VGPR0/1 hold K=k/k+1 (lanes 0-15 = cols).
// ---------------------------------------------------------------------------
template<int KIN>
__global__ void gemm_wmma_kernel(const float* __restrict__ X,
                                 const float* __restrict__ W,
                                 float* __restrict__ Y, int nTiles) {
  const int wid = (int)((blockIdx.x * blockDim.x + threadIdx.x) >> 5);
  if (wid >= nTiles) return;  // wave-uniform: full waves retire together
  const int lane = threadIdx.x & 31;
  const int hi   = lane >> 4;
  const int l16  = lane & 15;

  const float* xrow = X + (size_t)(wid * 16 + l16) * KIN;
  v8f c0 = {}, c1 = {}, c2 = {}, c3 = {};

#pragma unroll
  for (int kk = 0; kk < KIN; kk += 4) {
    const int k = kk + 2 * hi;
    v2f a;
    a.x = xrow[k];
    a.y = xrow[k + 1];
    const float* w0 = W + k * 64 + l16;
    const float* w1 = w0 + 64;
    v2f b;
    b.x = w0[0];  b.y = w1[0];
    c0 = __builtin_amdgcn_wmma_f32_16x16x4_f32(false, a, false, b, (short)0, c0, false, false);
    b.x = w0[16]; b.y = w1[16];
    c1 = __builtin_amdgcn_wmma_f32_16x16x4_f32(false, a, false, b, (short)0, c1, false, false);
    b.x = w0[32]; b.y = w1[32];
    c2 = __builtin_amdgcn_wmma_f32_16x16x4_f32(false, a, false, b, (short)0, c2, false, false);
    b.x = w0[48]; b.y = w1[48];
    c3 = __builtin_amdgcn_wmma_f32_16x16x4_f32(false, a, false, b, (short)0, c3, false, false);
  }

  // D layout: lane<16 -> row M=v, col N=l16 ; lane>=16 -> row M=v+8
  float* yb = Y + (size_t)(wid * 16 + 8 * hi) * 64 + l16;
#pragma unroll
  for (int v = 0; v < 8; ++v) {
    yb[v * 64 +  0] = c0[v];
    yb[v * 64 + 16] = c1[v];
    yb[v * 64 + 32] = c2[v];
    yb[v * 64 + 48] = c3[v];
  }
}

// ---------------------------------------------------------------------------
// Edge scatter: agg[col] += xw[row] * dinv[row]   (dinv[col] applied later)
// one thread = one edge x 4 features (float4 gather, 4 f32 atomics)
// ---------------------------------------------------------------------------
__global__ void scatter_kernel(const int* __restrict__ row, const int* __restrict__ col,
                               const float* __restrict__ xw, const float* __restrict__ dinv,
                               float* __restrict__ agg, int total) {
  int idx = blockIdx.x * blockDim.x + threadIdx.x;
  if (idx >= total) return;
  const int e = idx >> 4;
  const int f = (idx & 15) << 2;
  const int r = row[e], c = col[e];
  const float dr = dinv[r];
  const float4 v = *reinterpret_cast<const float4*>(xw + (size_t)r * 64 + f);
  float* dst = agg + (size_t)c * 64 + f;
  atomicAdd(dst + 0, v.x * dr);
  atomicAdd(dst + 1, v.y * dr);
  atomicAdd(dst + 2, v.z * dr);
  atomicAdd(dst + 3, v.w * dr);
}

// ---------------------------------------------------------------------------
// Finalize: y = dinv*agg + xw*dinv^2 + bias. If BN: write y into agg and
// accumulate per-feature sum/sumsq (LDS reduce + 1 atomic per feature per
// block). Else: write relu(y) to outNoBN.
// ---------------------------------------------------------------------------
template<bool BN>
__global__ void finalize_kernel(const float* __restrict__ xw, float* __restrict__ agg,
                                const float* __restrict__ dinv, const float* __restrict__ bias,
                                float* sum, float* sumsq, float* __restrict__ outNoBN,
                                int total) {
  __shared__ float s1[256];
  __shared__ float s2[256];
  const int f = threadIdx.x & 63;  // constant: stride is a multiple of 64
  float ls = 0.0f, lss = 0.0f;
  const float bf = bias[f];
  for (int idx = blockIdx.x * blockDim.x + threadIdx.x; idx < total;
       idx += gridDim.x * blockDim.x) {
    const int node = idx >> 6;
    const float di = dinv[node];
    const float y = di * agg[idx] + xw[idx] * di * di + bf;
    if (BN) {
      agg[idx] = y;
      ls += y;
      lss += y * y;
    } else {
      outNoBN[idx] = fmaxf(y, 0.0f);
    }
  }
  if (BN) {
    s1[threadIdx.x] = ls;
    s2[threadIdx.x] = lss;
    __syncthreads();
    if (threadIdx.x < 64) {
      float a = s1[threadIdx.x] + s1[threadIdx.x + 64] + s1[threadIdx.x + 128] + s1[threadIdx.x + 192];
      float b = s2[threadIdx.x] + s2[threadIdx.x + 64] + s2[threadIdx.x + 128] + s2[threadIdx.x + 192];
      atomicAdd(&sum[threadIdx.x], a);
      atomicAdd(&sumsq[threadIdx.x], b);
    }
  }
}

__global__ void bn_param_kernel(const float* sum, const float* sumsq,
                                const float* g, const float* be,
                                float* scale, float* shift, float invN) {
  const int f = threadIdx.x;  // 64 threads
  const float mu  = sum[f] * invN;
  const float var = sumsq[f] * invN - mu * mu;  // biased variance
  const float sc  = g[f] * rsqrtf(var + 1e-5f);
  scale[f] = sc;
  shift[f] = be[f] - mu * sc;
}

__global__ void bn_apply_kernel(const float* __restrict__ y, const float* __restrict__ scale,
                                const float* __restrict__ shift, float* __restrict__ xn,
                                int total) {
  int idx = blockIdx.x * blockDim.x + threadIdx.x;
  if (idx < total) {
    const int f = idx & 63;
    xn[idx] = fmaxf(y[idx] * scale[f] + shift[f], 0.0f);
  }
}

// ---------------------------------------------------------------------------
// Edge MLP: h = relu([ue|ie] @ fcW1 + fcb1); pred = sigmoid(h @ fcW2 + fcb2)
// One wave -> 16 edges. K=32 -> 8 WMMA steps x 4 col-tiles = 32 WMMA/wave.
// ---------------------------------------------------------------------------
__global__ void edge_mlp_kernel(const int* __restrict__ row, const int* __restrict__ col,
                                const float* __restrict__ ue, const float* __restrict__ ie,
                                const float* __restrict__ fcW1, const float* __restrict__ fcb1,
                                const float* __restrict__ fcW2, const float* __restrict__ fcb2,
                                float* __restrict__ pred, int nTiles) {
  const int wid = (int)((blockIdx.x * blockDim.x + threadIdx.x) >> 5);
  if (wid >= nTiles) return;
  const int lane = threadIdx.x & 31;
  const int hi   = lane >> 4;
  const int l16  = lane & 15;

  const int e = wid * 16 + l16;
  const float* up = ue + (size_t)row[e] * 16;
  const float* ip = ie + (size_t)(col[e] - NU_N) * 16;

  v8f c0 = {}, c1 = {}, c2 = {}, c3 = {};
#pragma unroll
  for (int j = 0; j < 8; ++j) {
    const int k = 4 * j + 2 * hi;                 // k even -> k,k+1 same side
    const float* ab = (k < 16) ? (up + k) : (ip + (k - 16));
    v2f a;
    a.x = ab[0];
    a.y = ab[1];
    const float* w0 = fcW1 + k * 64 + l16;
    const float* w1 = w0 + 64;
    v2f b;
    b.x = w0[0];  b.y = w1[0];
    c0 = __builtin_amdgcn_wmma_f32_16x16x4_f32(false, a, false, b, (short)0, c0, false, false);
    b.x = w0[16]; b.y = w1[16];
    c1 = __builtin_amdgcn_wmma_f32_16x16x4_f32(false, a, false, b, (short)0, c1, false, false);
    b.x = w0[32]; b.y = w1[32];
    c2 = __builtin_amdgcn_wmma_f32_16x16x4_f32(false, a, false, b, (short)0, c2, false, false);
    b.x = w0[48]; b.y = w1[48];
    c3 = __builtin_amdgcn_wmma_f32_16x16x4_f32(false, a, false, b, (short)0, c3, false, false);
  }

  // epilogue: relu + bias, dot with fcW2 per lane (each lane owns 4 columns)
  float part[8];
#pragma unroll
  for (int v = 0; v < 8; ++v) part[v] = 0.0f;
#pragma unroll
  for (int t = 0; t < 4; ++t) {
    const int n = t * 16 + l16;
    const float bias = fcb1[n];
    const float w2 = fcW2[n];
    const v8f cc = (t == 0) ? c0 : (t == 1) ? c1 : (t == 2) ? c2 : c3;
#pragma unroll
    for (int v = 0; v < 8; ++v)
      part[v] += fmaxf(cc[v] + bias, 0.0f) * w2;
  }
  // reduce across the 16-lane half (masks 1..8 never cross bit 4)
#pragma unroll
  for (int v = 0; v < 8; ++v) {
    part[v] += __shfl_xor(part[v], 1, 32);
    part[v] += __shfl_xor(part[v], 2, 32);
    part[v] += __shfl_xor(part[v], 4, 32);
    part[v] += __shfl_xor(part[v], 8, 32);
  }
  if (l16 == 0) {  // lane 0 -> edges m=0..7, lane 16 -> edges m=8..15
    const float b2 = fcb2[0];
    float* pd = pred + wid * 16 + hi * 8;
#pragma unroll
    for (int v = 0; v < 8; ++v) {
      const float z = part[v] + b2;
      pd[v] = 1.0f / (1.0f + __expf(-z));
    }
  }
}

// ---------------------------------------------------------------------------
extern "C" void kernel_launch(void* const* d_in, const int* in_sizes, int n_in,
                              void* d_out, int out_size, void* d_ws, size_t ws_size,
                              hipStream_t stream) {
  (void)in_sizes; (void)n_in; (void)out_size; (void)ws_size;
  const int*   edge     = (const int*)d_in[0];
  const int*   row      = edge;
  const int*   col      = edge + EE;
  const float* user_emb = (const float*)d_in[1];
  const float* item_emb = (const float*)d_in[2];
  const float* W1  = (const float*)d_in[3];
  const float* b1  = (const float*)d_in[4];
  const float* W2  = (const float*)d_in[5];
  const float* b2  = (const float*)d_in[6];
  const float* W3  = (const float*)d_in[7];
  const float* b3  = (const float*)d_in[8];
  const float* g1  = (const float*)d_in[9];
  const float* be1 = (const float*)d_in[10];
  const float* g2  = (const float*)d_in[11];
  const float* be2 = (const float*)d_in[12];
  const float* fcW1 = (const float*)d_in[13];
  const float* fcb1 = (const float*)d_in[14];
  const float* fcW2 = (const float*)d_in[15];
  const float* fcb2 = (const float*)d_in[16];

  float* ws    = (float*)d_ws;
  float* dinv  = ws;                         // [NN]
  float* sum   = ws + NN;                    // [64]
  float* sumsq = sum + 64;                   // [64]
  float* scale = sumsq + 64;                 // [64]
  float* shift = scale + 64;                 // [64]
  float* xcur  = ws + NN + 256;              // [NN*64]
  float* xw    = xcur + (size_t)NN * 64;     // [NN*64]
  float* agg   = xw + (size_t)NN * 64;       // [NN*64]

  float* pred = (float*)d_out;               // [EE]
  float* xout = (float*)d_out + EE;          // [NN*64]

  const int B = 256;
  const int NWORD = NN * 64;                 // 9,600,000
  const int SCAT  = EE * 16;                 // 32,000,000
  auto cdiv = [](int a, int b) { return (a + b - 1) / b; };

  // degrees -> dinv (graph is fixed across all three layers)
  init_deg_kernel<<<cdiv(NN, B), B, 0, stream>>>(dinv, NN);
  deg_count_kernel<<<cdiv(EE, B), B, 0, stream>>>(col, dinv, EE);
  deg_rsqrt_kernel<<<cdiv(NN, B), B, 0, stream>>>(dinv, NN);

  // -------- layer 1: x = relu(BN(gcn(concat(ue,ie), W1, b1)))
  gemm_wmma_kernel<16><<<cdiv(NU_N / 16, 8), B, 0, stream>>>(user_emb, W1, xw, NU_N / 16);
  gemm_wmma_kernel<16><<<cdiv(NI_N / 16, 8), B, 0, stream>>>(item_emb, W1,
                                                             xw + (size_t)NU_N * 64, NI_N / 16);
  zero_kernel<<<cdiv(NWORD, B), B, 0, stream>>>(agg, NWORD);
  zero_kernel<<<1, 128, 0, stream>>>(sum, 128);
  scatter_kernel<<<cdiv(SCAT, B), B, 0, stream>>>(row, col, xw, dinv, agg, SCAT);
  finalize_kernel<true><<<2048, B, 0, stream>>>(xw, agg, dinv, b1, sum, sumsq, nullptr, NWORD);
  bn_param_kernel<<<1, 64, 0, stream>>>(sum, sumsq, g1, be1, scale, shift, 1.0f / NN);
  bn_apply_kernel<<<cdiv(NWORD, B), B, 0, stream>>>(agg, scale, shift, xcur, NWORD);

  // -------- layer 2
  gemm_wmma_kernel<64><<<cdiv(NN / 16, 8), B, 0, stream>>>(xcur, W2, xw, NN / 16);
  zero_kernel<<<cdiv(NWORD, B), B, 0, stream>>>(agg, NWORD);
  zero_kernel<<<1, 128, 0, stream>>>(sum, 128);
  scatter_kernel<<<cdiv(SCAT, B), B, 0, stream>>>(row, col, xw, dinv, agg, SCAT);
  finalize_kernel<true><<<2048, B, 0, stream>>>(xw, agg, dinv, b2, sum, sumsq, nullptr, NWORD);
  bn_param_kernel<<<1, 64, 0, stream>>>(sum, sumsq, g2, be2, scale, shift, 1.0f / NN);
  bn_apply_kernel<<<cdiv(NWORD, B), B, 0, stream>>>(agg, scale, shift, xcur, NWORD);

  // -------- layer 3 (no BN): relu -> d_out node block
  gemm_wmma_kernel<64><<<cdiv(NN / 16, 8), B, 0, stream>>>(xcur, W3, xw, NN / 16);
  zero_kernel<<<cdiv(NWORD, B), B, 0, stream>>>(agg, NWORD);
  scatter_kernel<<<cdiv(SCAT, B), B, 0, stream>>>(row, col, xw, dinv, agg, SCAT);
  finalize_kernel<false><<<2048, B, 0, stream>>>(xw, agg, dinv, b3, nullptr, nullptr, xout, NWORD);

  // -------- edge MLP (independent of GCN stack)
  edge_mlp_kernel<<<cdiv(EE / 16, 8), B, 0, stream>>>(row, col, user_emb, item_emb,
                                                      fcW1, fcb1, fcW2, fcb2, pred, EE / 16);
}